// RelRepWindowContext_2834678415840
// MI455X (gfx1250) — compile-verified
//
// RelRepWindowContext fused kernel for MI455X (gfx1250, CDNA5, wave32).
//
// Strategy: compute-bound fused span-pair MLP using V_WMMA_F32_16X16X4_F32
// (exact fp32 accumulate, matching the fp32 reference numerics).
//   kernel 1: windowed max-pool ctx[B,K,H] -> workspace
//   kernel 2: per (b,i) workgroup, build A2=[span_j | max(ctx_i,ctx_j)] in LDS,
//             hoist span_i * W1[0:256,:] into vec0 (shared by all 64 rows),
//             then stream FFN in 64-col chunks: GEMM1(WMMA)->relu->LDS,
//             GEMM2(WMMA) accumulates the 64x256 output tile in registers.
#include <hip/hip_runtime.h>
#include <stdint.h>

#define B_   8
#define K_   64
#define L_   1024
#define H_   256
#define FFN_ 3072
#define WIN_ 20

#define A2S  516   // A2 row stride (floats); 516 % 64 == 4 -> conflict-free b64 frag loads
#define HCS  68    // Hc row stride;          68 % 64 == 4
#define NC   64    // FFN columns per fused chunk

typedef float v2f __attribute__((ext_vector_type(2)));
typedef float v8f __attribute__((ext_vector_type(8)));

__device__ __forceinline__ v8f v8f_zero() {
  v8f z;
#pragma unroll
  for (int p = 0; p < 8; ++p) z[p] = 0.0f;
  return z;
}

// D = A(16x4 f32) * B(4x16 f32) + C(16x16 f32), full fp32 WMMA.
__device__ __forceinline__ v8f wmma4(v2f a, v2f b, v8f c) {
  return __builtin_amdgcn_wmma_f32_16x16x4_f32(
      /*neg_a=*/false, a, /*neg_b=*/false, b,
      /*c_mod=*/(short)0, c, /*reuse_a=*/false, /*reuse_b=*/false);
}

// ---------------- Kernel 1: per-span window max over tokens ----------------
__global__ __launch_bounds__(256) void ctx_window_max_kernel(
    const float* __restrict__ token_reps,      // [B,L,H]
    const int* __restrict__ span_ids,          // [B,K,2]
    const unsigned char* __restrict__ token_masks,  // [B,L] (bool)
    float* __restrict__ ctx) {                 // [B,K,H]
  const int bk = blockIdx.x;            // b*K + k
  const int b  = bk >> 6;
  const int h  = threadIdx.x;           // one channel per thread (H=256)
  const int s  = span_ids[2 * bk + 0];
  const int e  = span_ids[2 * bk + 1];
  const float* tr = token_reps + (size_t)b * L_ * H_;
  const unsigned char* mk = token_masks + (size_t)b * L_;
  float m = -INFINITY;
  int lo = s - WIN_; if (lo < 0) lo = 0;
  for (int t = lo; t < s; ++t)                    // [start-W, start)
    if (mk[t]) m = fmaxf(m, tr[(size_t)t * H_ + h]);
  int hi = e + WIN_; if (hi > L_ - 1) hi = L_ - 1;
  for (int t = e + 1; t <= hi; ++t)               // (end, end+W]
    if (mk[t]) m = fmaxf(m, tr[(size_t)t * H_ + h]);
  ctx[(size_t)bk * H_ + h] = m;
}

// ---------------- Kernel 2: fused pair-MLP, one WG per (b,i) ----------------
__global__ __launch_bounds__(256) void relrep_mlp_kernel(
    const float* __restrict__ span_reps,  // [B,K,H]
    const float* __restrict__ ctx,        // [B,K,H]
    const float* __restrict__ W1,         // [768, 3072] row-major
    const float* __restrict__ b1,         // [3072]
    const float* __restrict__ W2,         // [3072, 256]
    const float* __restrict__ b2,         // [256]
    float* __restrict__ out) {            // [B, K*K, H]
  __shared__ __align__(16) float sA2[K_ * A2S];   // [span_j | max(ctx_i,ctx_j)]
  __shared__ __align__(16) float sHc[K_ * HCS];   // relu(h) chunk staging
  __shared__ __align__(16) float sV0[FFN_];       // b1 + span_i @ W1[0:256,:]

  const int tid  = threadIdx.x;
  const int lane = tid & 31;
  const int wv   = tid >> 5;        // 8 waves
  const int half = lane >> 4;       // lane group 0..15 / 16..31
  const int mn   = lane & 15;       // M (A rows / C cols) or N index

  const int b = blockIdx.x >> 6;
  const int i = blockIdx.x & 63;

  const float* spans = span_reps + (size_t)b * K_ * H_;
  const float* ctxb  = ctx       + (size_t)b * K_ * H_;

  // Build A2[j] = [ span_j (256) | max(ctx_i, ctx_j) (256) ]
  for (int idx = tid; idx < K_ * H_; idx += 256) {
    const int j = idx >> 8;
    const int c = idx & 255;
    sA2[j * A2S + c]      = spans[idx];
    sA2[j * A2S + H_ + c] = fmaxf(ctxb[i * H_ + c], ctxb[idx]);
  }
  __syncthreads();

  // vec0[f] = b1[f] + sum_c span_i[c] * W1[c][f]  (span_i identical for all 64 rows)
  for (int f = tid; f < FFN_; f += 256) {
    float s = b1[f];
    const float* sp = &sA2[i * A2S];
#pragma unroll 8
    for (int c = 0; c < H_; ++c)
      s = fmaf(sp[c], W1[(size_t)c * FFN_ + f], s);
    sV0[f] = s;
  }
  __syncthreads();

  // Output accumulators: wave wv owns out cols [32*wv, 32*wv+32), all 4 row tiles.
  v8f oacc[4][2];
#pragma unroll
  for (int rt = 0; rt < 4; ++rt) {
    oacc[rt][0] = v8f_zero();
    oacc[rt][1] = v8f_zero();
  }

  // GEMM1 tile ownership: wave wv -> col-tile (wv&3), row tiles {2*(wv>>2), 2*(wv>>2)+1}
  const int g1_ct  = wv & 3;
  const int g1_rt0 = (wv >> 2) * 2;

  for (int c0 = 0; c0 < FFN_; c0 += NC) {
    // ---- GEMM1: Hc = relu( A2[64,512] @ W1[256:768, c0:c0+NC] + vec0 ) ----
    v8f hacc0 = v8f_zero();
    v8f hacc1 = v8f_zero();
    // B fragment base: W1 row 256 + 2*half (+v via +FFN_), col c0 + ct*16 + mn
    const float* w1base =
        W1 + ((size_t)H_ + 2 * half) * FFN_ + c0 + g1_ct * 16 + mn;
    const float* a0base = &sA2[(g1_rt0 * 16 + mn) * A2S + 2 * half];
    const float* a1base = &sA2[((g1_rt0 + 1) * 16 + mn) * A2S + 2 * half];
#pragma unroll 4
    for (int kk = 0; kk < 512; kk += 4) {
      const float* wp = w1base + (size_t)kk * FFN_;
      v2f bf; bf.x = wp[0]; bf.y = wp[FFN_];
      v2f a0 = *(const v2f*)(a0base + kk);
      v2f a1 = *(const v2f*)(a1base + kk);
      hacc0 = wmma4(a0, bf, hacc0);
      hacc1 = wmma4(a1, bf, hacc1);
    }
    {
      const float v0 = sV0[c0 + g1_ct * 16 + mn];
      const int colo = g1_ct * 16 + mn;
#pragma unroll
      for (int p = 0; p < 8; ++p) {
        const int r0 = g1_rt0 * 16 + p + 8 * half;       // C/D layout: M = p + 8*half
        const int r1 = (g1_rt0 + 1) * 16 + p + 8 * half;
        sHc[r0 * HCS + colo] = fmaxf(hacc0[p] + v0, 0.0f);
        sHc[r1 * HCS + colo] = fmaxf(hacc1[p] + v0, 0.0f);
      }
    }
    __syncthreads();

    // ---- GEMM2: oacc += Hc[64,NC] @ W2[c0:c0+NC, 32*wv : 32*wv+32] ----
    const float* w2base = W2 + ((size_t)c0 + 2 * half) * H_ + 32 * wv + mn;
#pragma unroll 2
    for (int kk = 0; kk < NC; kk += 4) {
      const float* wp = w2base + (size_t)kk * H_;
      v2f bf0; bf0.x = wp[0];  bf0.y = wp[H_];
      v2f bf1; bf1.x = wp[16]; bf1.y = wp[16 + H_];
#pragma unroll
      for (int rt = 0; rt < 4; ++rt) {
        v2f af = *(const v2f*)&sHc[(rt * 16 + mn) * HCS + kk + 2 * half];
        oacc[rt][0] = wmma4(af, bf0, oacc[rt][0]);
        oacc[rt][1] = wmma4(af, bf1, oacc[rt][1]);
      }
    }
    __syncthreads();  // Hc reused next chunk
  }

  // Store: out[b, i*K + j, h] = oacc + b2[h]
  const float bb0 = b2[32 * wv + mn];
  const float bb1 = b2[32 * wv + 16 + mn];
  float* ob = out + ((size_t)b * (K_ * K_) + (size_t)i * K_) * H_;
#pragma unroll
  for (int rt = 0; rt < 4; ++rt) {
#pragma unroll
    for (int p = 0; p < 8; ++p) {
      const int j = rt * 16 + p + 8 * half;
      ob[(size_t)j * H_ + 32 * wv + mn]      = oacc[rt][0][p] + bb0;
      ob[(size_t)j * H_ + 32 * wv + 16 + mn] = oacc[rt][1][p] + bb1;
    }
  }
}

extern "C" void kernel_launch(void* const* d_in, const int* in_sizes, int n_in,
                              void* d_out, int out_size, void* d_ws, size_t ws_size,
                              hipStream_t stream) {
  (void)in_sizes; (void)n_in; (void)out_size; (void)ws_size;
  const float* span_reps   = (const float*)d_in[0];
  const float* token_reps  = (const float*)d_in[1];
  const float* W1          = (const float*)d_in[2];
  const float* b1          = (const float*)d_in[3];
  const float* W2          = (const float*)d_in[4];
  const float* b2          = (const float*)d_in[5];
  const int*   span_ids    = (const int*)d_in[6];
  const unsigned char* tmk = (const unsigned char*)d_in[7];
  float* out = (float*)d_out;
  float* ctx = (float*)d_ws;  // [B,K,H] = 512 KB scratch

  ctx_window_max_kernel<<<dim3(B_ * K_), dim3(H_), 0, stream>>>(
      token_reps, span_ids, tmk, ctx);
  relrep_mlp_kernel<<<dim3(B_ * K_), dim3(256), 0, stream>>>(
      span_reps, ctx, W1, b1, W2, b2, out);
}